// SparseStem_69965017252011
// MI455X (gfx1250) — compile-verified
//
#include <hip/hip_runtime.h>

typedef __attribute__((ext_vector_type(2))) float v2f;
typedef __attribute__((ext_vector_type(8))) float v8f;

#define B_    2
#define D_    12
#define H_    200
#define W_    200
#define CELLS 960000
#define NPTS  60000
#define COUT_ 64
#define EPS_  1e-5f

// padded (halo) grid: (B, D+2, H+2, W+2)
#define PD_   (D_ + 2)
#define PW_   (W_ + 2)
#define PH_   (H_ + 2)
#define PHW_  (PH_ * PW_)
#define PCELLS (B_ * PD_ * PHW_)          // 1,142,512

// per-tap geometry (tap t: fz=t/9, fy=(t/3)%3, fx=t%3; d* = f* - 1)
constexpr int kDZ[27] = {-1,-1,-1,-1,-1,-1,-1,-1,-1, 0,0,0,0,0,0,0,0,0, 1,1,1,1,1,1,1,1,1};
constexpr int kDY[27] = {-1,-1,-1, 0,0,0, 1,1,1, -1,-1,-1, 0,0,0, 1,1,1, -1,-1,-1, 0,0,0, 1,1,1};
constexpr int kDX[27] = {-1,0,1, -1,0,1, -1,0,1, -1,0,1, -1,0,1, -1,0,1, -1,0,1, -1,0,1, -1,0,1};
#define POFF(t) (kDZ[t] * PHW_ + kDY[t] * PW_ + kDX[t])   // compile-time

// ---------------------------------------------------------------------------
// Pass 1a: padded map defaults to sentinel NPTS (-> zero feature row)
// ---------------------------------------------------------------------------
__global__ void init_pmap_kernel(int* __restrict__ pmap) {
    int i = blockIdx.x * blockDim.x + threadIdx.x;
    if (i < PCELLS) pmap[i] = NPTS;
}

// ---------------------------------------------------------------------------
// Pass 1b: pmap[padded_cell_of_point_n] = n.  Tap 13 = (1,1,1) is the
// identity tap, so scatter_idx[13*N + n] is point n's own cell.  Points
// occupy distinct cells -> no write race.
// ---------------------------------------------------------------------------
__global__ void fill_pmap_kernel(const int* __restrict__ scatter_idx,
                                 int* __restrict__ pmap) {
    int n = blockIdx.x * blockDim.x + threadIdx.x;
    if (n < NPTS) {
        int c = scatter_idx[13 * NPTS + n];
        int x = c % W_;  c /= W_;
        int y = c % H_;  c /= H_;
        int z = c % D_;
        int b = c / D_;
        int pc = ((b * PD_ + z + 1) * PH_ + y + 1) * PW_ + x + 1;
        pmap[pc] = n;
    }
}

// ---------------------------------------------------------------------------
// Pass 1c: features copied with one extra zero row at index NPTS
// ---------------------------------------------------------------------------
__global__ void copy_feat_kernel(const float* __restrict__ features,
                                 float* __restrict__ fbuf) {
    int i = blockIdx.x * blockDim.x + threadIdx.x;
    if (i < 2 * NPTS + 2) fbuf[i] = (i < 2 * NPTS) ? features[i] : 0.0f;
}

// ---------------------------------------------------------------------------
// Fused gather-conv (WMMA f32 16x16x4) + LayerNorm + ReLU + active mask.
// One wave32 per 16-cell tile; 8 waves (256 threads) per block.
//
// GEMM per tile: acc(16x64) = A(16x56) x B(56x64).  K remap: K = 4kk+2h+v
// <-> tap = 2kk+v, cin p = h (lane half).  Per K-step each lane gathers two
// taps' single-cin features; tap offsets are compile-time constants into the
// halo-padded map, so the loop has NO bounds checks and NO selects: a miss
// returns sentinel NPTS which indexes the zeroed feature row.  Map (4.6MB)
// and features (0.5MB) are L2-resident (192MB L2); the only HBM-bound
// traffic is the single 246MB output stream (~10.5us floor at 23.3TB/s).
//
// LDS B layout [p][col][tap]: fragment pairs (tap 2kk, 2kk+1) are contiguous
// 8B -> ds_load_b64 with immediate offsets off one per-lane base.
//
// A 5th accumulator c4 = A x wsum (wsum[K] = sum_n B[K,n]) puts each row's
// channel-sum directly in-lane (all B columns equal) -> LN mean needs no
// cross-lane reduction; only the second moment uses 4 shfl_xor steps.
// ---------------------------------------------------------------------------
__global__ __launch_bounds__(256) void conv_ln_kernel(
    const float* __restrict__ fbuf,       // (N+1, 2), row N = zeros
    const float* __restrict__ weight,     // (27, 2, 64)
    const float* __restrict__ gamma,      // (64)
    const float* __restrict__ beta,       // (64)
    const int*   __restrict__ pmap,       // (PCELLS)
    float*       __restrict__ out)        // (CELLS, 64)
{
    __shared__ float Wlds[2 * 64 * 28];   // [p][col][tap], tap 27 zeroed
    __shared__ float WsumLds[2 * 28];     // [p][tap] row sums, tap 27 zero

    // transpose-fill: Wlds[(p*64+col)*28+tap] = weight[(tap*2+p)*64+col]
    for (int i = threadIdx.x; i < 2 * 64 * 28; i += 256) {
        const int tap = i % 28;
        const int col = (i / 28) & 63;
        const int p   = i / (28 * 64);
        Wlds[i] = (tap < 27) ? weight[(tap * 2 + p) * 64 + col] : 0.0f;
    }
    __syncthreads();
    if (threadIdx.x < 56) {
        const int tap = threadIdx.x % 28;
        const int p   = threadIdx.x / 28;
        float s = 0.0f;
#pragma unroll
        for (int q = 0; q < 64; ++q) s += Wlds[(p * 64 + q) * 28 + tap];
        WsumLds[p * 28 + tap] = s;
    }
    __syncthreads();

    const int lane = threadIdx.x & 31;
    const int wave = threadIdx.x >> 5;
    const int m    = lane & 15;           // N-column within tile / cell row
    const int half = lane >> 4;           // = cin index p
    const int cellBase = (blockIdx.x * 8 + wave) * 16;
    const int cell = cellBase + m;

    // decode cell -> padded-grid center index (neighbors = pcell + const)
    int xl = cell % W_;
    int t  = cell / W_;
    int yl = t % H_;
    t /= H_;
    int zl = t % D_;
    int bl = t / D_;
    const int pcell = ((bl * PD_ + zl + 1) * PH_ + yl + 1) * PW_ + xl + 1;

    float g[4], bt[4];
#pragma unroll
    for (int j = 0; j < 4; ++j) {
        g[j]  = gamma[16 * j + m];
        bt[j] = beta [16 * j + m];
    }

    // per-lane LDS bases; all step/tile offsets become DS immediates
    const float* __restrict__ wbase  = &Wlds[(half * 64 + m) * 28];  // +j*448+kk*2
    const float* __restrict__ wsbase = &WsumLds[half * 28];          // +kk*2

    v8f c0 = {}, c1 = {}, c2 = {}, c3 = {}, c4 = {};
    int nmin = NPTS;                      // < NPTS  <=>  at least one hit

#pragma unroll
    for (int kk = 0; kk < 14; ++kk) {
        const int t0 = 2 * kk, t1 = t0 + 1;

        const int i0 = pmap[pcell + POFF(t0)];         // L2-resident
        nmin = (i0 < nmin) ? i0 : nmin;
        const float a0 = fbuf[i0 * 2 + half];          // miss -> zero row

        float a1 = 0.0f;
        if (t1 < 27) {
            const int i1 = pmap[pcell + POFF(t1)];
            nmin = (i1 < nmin) ? i1 : nmin;
            a1 = fbuf[i1 * 2 + half];
        }
        const v2f a = { a0, a1 };

        // B-frags: contiguous (tap even, tap odd) pairs -> ds_load_b64
        const v2f b0 = *(const v2f*)&wbase[kk * 2 +    0];
        const v2f b1 = *(const v2f*)&wbase[kk * 2 +  448];
        const v2f b2 = *(const v2f*)&wbase[kk * 2 +  896];
        const v2f b3 = *(const v2f*)&wbase[kk * 2 + 1344];
        const v2f bs = *(const v2f*)&wsbase[kk * 2];

        c0 = __builtin_amdgcn_wmma_f32_16x16x4_f32(false, a, false, b0, (short)0, c0, false, false);
        c1 = __builtin_amdgcn_wmma_f32_16x16x4_f32(false, a, false, b1, (short)0, c1, false, false);
        c2 = __builtin_amdgcn_wmma_f32_16x16x4_f32(false, a, false, b2, (short)0, c2, false, false);
        c3 = __builtin_amdgcn_wmma_f32_16x16x4_f32(false, a, false, b3, (short)0, c3, false, false);
        c4 = __builtin_amdgcn_wmma_f32_16x16x4_f32(false, a, false, bs, (short)0, c4, false, false);
    }

    // both halves saw all 27 taps of their cell -> one mask bit per cell row
    const unsigned mask = (unsigned)__ballot(nmin < NPTS);

#pragma unroll
    for (int v = 0; v < 8; ++v) {
        // second moment across the 16 columns held by this half-wave
        float s2 = c0[v]*c0[v] + c1[v]*c1[v] + c2[v]*c2[v] + c3[v]*c3[v];
#pragma unroll
        for (int off = 1; off < 16; off <<= 1)
            s2 += __shfl_xor(s2, off, 32);

        const float mean  = c4[v] * (1.0f / 64.0f);    // row sum via WMMA
        const float var   = s2 * (1.0f / 64.0f) - mean * mean;
        const float rinv  = rsqrtf(var + EPS_);
        const float mrinv = -mean * rinv;

        const int  mr  = v + 8 * half;
        const bool act = ((mask >> mr) & 1u) != 0;

        float* op = out + (size_t)(cellBase + mr) * COUT_ + m;
        const float vals[4] = { c0[v], c1[v], c2[v], c3[v] };
#pragma unroll
        for (int j = 0; j < 4; ++j) {
            const float nrm = vals[j] * rinv + mrinv;  // (x-mu)*rsqrt
            float o = nrm * g[j] + bt[j];
            o = fmaxf(o, 0.0f);
            op[16 * j] = act ? o : 0.0f;               // 64B runs per half-wave
        }
    }
}

// ---------------------------------------------------------------------------
extern "C" void kernel_launch(void* const* d_in, const int* in_sizes, int n_in,
                              void* d_out, int out_size, void* d_ws, size_t ws_size,
                              hipStream_t stream) {
    const float* features = (const float*)d_in[0];   // (N,2)
    const float* weight   = (const float*)d_in[1];   // (27,2,64)
    const float* gamma    = (const float*)d_in[2];   // (64)
    const float* beta     = (const float*)d_in[3];   // (64)
    const int*   scatter  = (const int*)  d_in[4];   // (27,N)
    float* out = (float*)d_out;

    int*   pmap = (int*)d_ws;                        // 4.57 MB (L2-resident)
    float* fbuf = (float*)((char*)d_ws + (size_t)PCELLS * sizeof(int)); // 0.48 MB

    init_pmap_kernel<<<(PCELLS + 255) / 256, 256, 0, stream>>>(pmap);
    fill_pmap_kernel<<<(NPTS + 255) / 256, 256, 0, stream>>>(scatter, pmap);
    copy_feat_kernel<<<(2 * NPTS + 2 + 255) / 256, 256, 0, stream>>>(features, fbuf);
    conv_ln_kernel<<<CELLS / (16 * 8), 256, 0, stream>>>(
        fbuf, weight, gamma, beta, pmap, out);
}